// GroupingLayer_58136677318895
// MI455X (gfx1250) — compile-verified
//
#include <hip/hip_runtime.h>

// ---------------------------------------------------------------------------
// GroupingLayer (PointNet++ ball query + grouping) for MI455X / gfx1250.
//
// B=8, N=16384, M=2048, C=128, K=64, R=0.2.
// Memory-bound: ~549 MB output stream + gathers served from L2 (features =
// 64 MB < 192 MB L2).  Distance math uses V_WMMA_F32_16X16X4_F32 (fp32, K=4
// GEMM — exactly the coordinate-matmul shape CDNA5 provides):
//   D[m][n] = ||p_n||^2 - 2 c_m . p_n   (A=[cx,cy,cz,0], B=[-2p,0], C=||p||^2)
//   in-ball  <=>  D[m][n] < r^2 - ||c_m||^2   (per-row threshold)
// ---------------------------------------------------------------------------

typedef __attribute__((ext_vector_type(2))) float v2f;
typedef __attribute__((ext_vector_type(8))) float v8f;

#define BATCH 8
#define NPTS  16384
#define MCEN  2048
#define CFEAT 128
#define KCAP  64
#define R2    0.04f
#define INV_R 5.0f

// ---------------------------------------------------------------------------
// Kernel 1: ball query.  One wave32 per tile of 16 centroids.  Scans the N
// points 16 at a time with one fp32 WMMA per chunk; ballots give per-row
// masks; 16 lanes compact their own row's indices (ascending => matches the
// reference's stable-sort order).  Uniform early exit when all rows have K.
// Point loads are software-pipelined one chunk ahead so the s_wait_loadcnt
// overlaps a full iteration of WMMA/ballot/compaction work.
// ---------------------------------------------------------------------------
__global__ __launch_bounds__(32) void ball_query_wmma(
    const float* __restrict__ pts,    // (B, N, 3)
    const float* __restrict__ cents,  // (B, M, 3)
    int* __restrict__ idxbuf)         // (B, M, K) in d_ws
{
  const int lane = threadIdx.x;                    // 0..31
  const int tile = blockIdx.x;                     // 0 .. B*M/16 - 1
  const int b  = tile / (MCEN / 16);
  const int m0 = (tile % (MCEN / 16)) * 16;

  const float* cb = cents + ((size_t)b * MCEN + m0) * 3;
  const float* pb = pts   + (size_t)b * NPTS * 3;

  // A matrix 16x4 f32: row m = [cx, cy, cz, 0].
  // ISA layout: lanes 0-15 hold A[m][0..1] in vgpr0/1, lanes 16-31 hold
  // A[m][2..3] (m = lane%16).
  const int mrow = lane & 15;
  const float cax = cb[mrow * 3 + 0];
  const float cay = cb[mrow * 3 + 1];
  const float caz = cb[mrow * 3 + 2];
  v2f amat;
  amat.x = (lane < 16) ? cax : caz;
  amat.y = (lane < 16) ? cay : 0.0f;

  // Per-row thresholds th[i] = r^2 - ||c||^2 for C/D row (i + 8*(lane>=16)).
  float th[8];
#pragma unroll
  for (int i = 0; i < 8; ++i) {
    const int r = (lane < 16) ? i : (i + 8);
    const float x = cb[r * 3 + 0];
    const float y = cb[r * 3 + 1];
    const float z = cb[r * 3 + 2];
    th[i] = R2 - (x * x + y * y + z * z);
  }

  int cnt = (lane < 16) ? 0 : KCAP;  // idle lanes pre-satisfy the exit test
  int* myrow = idxbuf + ((size_t)(b * MCEN + m0 + (lane & 15))) * KCAP;

  // Prologue: load chunk 0's point for this lane's column.
  const int ncol = lane & 15;
  const float* pr0 = pb + (size_t)ncol * 3;
  float px = pr0[0], py = pr0[1], pz = pr0[2];

  for (int nb = 0; nb < NPTS; nb += 16) {
    // Consume the pipelined point, then immediately issue next chunk's load.
    const float cpx = px, cpy = py, cpz = pz;
    if (nb + 16 < NPTS) {
      const float* q = pb + (size_t)(nb + 16 + ncol) * 3;
      px = q[0]; py = q[1]; pz = q[2];
    }

    // B matrix 4x16 f32: col n = [-2px, -2py, -2pz, 0].
    // Layout mirror of A: lanes 0-15 hold rows K=0..1, lanes 16-31 rows 2..3.
    const float pn2 = cpx * cpx + cpy * cpy + cpz * cpz;
    v2f bmat;
    bmat.x = (lane < 16) ? (-2.0f * cpx) : (-2.0f * cpz);
    bmat.y = (lane < 16) ? (-2.0f * cpy) : 0.0f;

    // C matrix broadcast of ||p_n||^2 (col = lane%16 in both halves).
    v8f cacc;
#pragma unroll
    for (int i = 0; i < 8; ++i) cacc[i] = pn2;

    // D = A*B + C = ||p||^2 - 2 c.p   (16x16 tile, 8 VGPRs)
    v8f d = __builtin_amdgcn_wmma_f32_16x16x4_f32(
        false, amat, false, bmat, (short)0, cacc, false, false);

    // One ballot per D VGPR: bits[15:0] = row i cols 0..15,
    // bits[31:16] = row i+8 cols 0..15.
    unsigned bal[8];
#pragma unroll
    for (int i = 0; i < 8; ++i)
      bal[i] = __builtin_amdgcn_ballot_w32(d[i] < th[i]);

    // Compaction: lane L (<16) owns centroid row L of the tile.
    unsigned bv = 0;
    const unsigned bsel = (unsigned)(lane & 7);
#pragma unroll
    for (int i = 0; i < 8; ++i) bv = (bsel == (unsigned)i) ? bal[i] : bv;
    unsigned m16 = (lane < 8) ? (bv & 0xFFFFu) : (bv >> 16);
    if (lane >= 16) m16 = 0;

    while (m16 && cnt < KCAP) {          // ascending bit = ascending index
      const int bit = __ffs(m16) - 1;
      myrow[cnt] = nb + bit;
      ++cnt;
      m16 &= m16 - 1;
    }

    // Uniform early exit once every row has K candidates.
    if (__builtin_amdgcn_ballot_w32(cnt >= KCAP) == 0xFFFFFFFFu) break;
  }

  if (lane < 16) {
    for (int j = cnt; j < KCAP; ++j) myrow[j] = 0;  // zero-pad like reference
  }
}

// ---------------------------------------------------------------------------
// Kernel 2: gather + concat.  One 64-thread block per (b, m); lane = k.
// Writes are coalesced 256B non-temporal streams (keep the 549 MB output out
// of L2 so features stay resident); feature gathers hit L2 (64 MB tensor).
// ---------------------------------------------------------------------------
__global__ __launch_bounds__(64) void group_gather(
    const float* __restrict__ pts,    // (B, N, 3)
    const float* __restrict__ cents,  // (B, M, 3)
    const float* __restrict__ feat,   // (B, C, N)
    const int* __restrict__ idxbuf,   // (B, M, K)
    float* __restrict__ out)          // (B, C+3, M, K)
{
  const int k  = threadIdx.x;           // 0..K-1
  const int bm = blockIdx.x;            // 0..B*M-1
  const int b  = bm >> 11;              // / MCEN
  const int m  = bm & (MCEN - 1);

  const int id = idxbuf[(size_t)bm * KCAP + k];

  const float* p = pts + ((size_t)b * NPTS + id) * 3;
  const float* c = cents + (size_t)bm * 3;
  const float gx = (p[0] - c[0]) * INV_R;
  const float gy = (p[1] - c[1]) * INV_R;
  const float gz = (p[2] - c[2]) * INV_R;

  const size_t chstride = (size_t)MCEN * KCAP;
  float* ob = out + (((size_t)b * (CFEAT + 3)) * MCEN + m) * KCAP + k;
  __builtin_nontemporal_store(gx, ob);
  __builtin_nontemporal_store(gy, ob + chstride);
  __builtin_nontemporal_store(gz, ob + 2 * chstride);

  const float* fb = feat + (size_t)b * CFEAT * NPTS + id;
  float* oc = ob + 3 * chstride;
#pragma unroll 8
  for (int ch = 0; ch < CFEAT; ++ch) {
    __builtin_nontemporal_store(fb[(size_t)ch * NPTS], oc);
    oc += chstride;
  }
}

// ---------------------------------------------------------------------------
extern "C" void kernel_launch(void* const* d_in, const int* in_sizes, int n_in,
                              void* d_out, int out_size, void* d_ws,
                              size_t ws_size, hipStream_t stream) {
  const float* pts   = (const float*)d_in[0];  // (8, 16384, 3)
  const float* cents = (const float*)d_in[1];  // (8, 2048, 3)
  const float* feat  = (const float*)d_in[2];  // (8, 128, 16384)
  float* out = (float*)d_out;                  // (8, 131, 2048, 64)
  int* idxbuf = (int*)d_ws;                    // 8*2048*64 ints = 4 MB scratch

  ball_query_wmma<<<BATCH * (MCEN / 16), 32, 0, stream>>>(pts, cents, idxbuf);
  group_gather<<<BATCH * MCEN, 64, 0, stream>>>(pts, cents, feat, idxbuf, out);
}